// DTW_71236327571899
// MI455X (gfx1250) — compile-verified
//
#include <hip/hip_runtime.h>
#include <hip/hip_bf16.h>
#include <math.h>

#define HN   4096          // sequence length (H == W)
#define FEAT 16            // feature dim
#define NT   1024          // DP threads (one workgroup)
#define RPT  4             // rows per DP thread (NT*RPT == HN)

typedef __attribute__((ext_vector_type(2))) float v2f;
typedef __attribute__((ext_vector_type(8))) float v8f;

// ---------------------------------------------------------------------------
// Kernel 0: row norms  ny[i] = sum_k y[i][k]^2 ,  nyh[i] = sum_k yh[i][k]^2
// ---------------------------------------------------------------------------
__global__ void dtw_norms(const float* __restrict__ y,
                          const float* __restrict__ yh,
                          float* __restrict__ ny,
                          float* __restrict__ nyh) {
    int i = blockIdx.x * blockDim.x + threadIdx.x;
    if (i < HN) {
        float sa = 0.f, sb = 0.f;
#pragma unroll
        for (int k = 0; k < FEAT; ++k) {
            float a = y[i * FEAT + k];
            float b = yh[i * FEAT + k];
            sa = fmaf(a, a, sa);
            sb = fmaf(b, b, sb);
        }
        ny[i]  = sa;
        nyh[i] = sb;
    }
}

// ---------------------------------------------------------------------------
// Kernel 1: D[i][j] = (ny[i] + nyh[j] - 2 * y_i . yh_j) / 16 via WMMA f32.
// One wave per 16x16 tile of D; K=16 done as 4 chained V_WMMA_F32_16X16X4_F32.
// A operand (16x4 f32): lanes 0-15 carry K={4c,4c+1} of row M=lane,
//                       lanes 16-31 carry K={4c+2,4c+3}.  B maps identically.
// ---------------------------------------------------------------------------
__global__ __launch_bounds__(256)
void dtw_dist_wmma(const float* __restrict__ y,
                   const float* __restrict__ yh,
                   const float* __restrict__ ny,
                   const float* __restrict__ nyh,
                   float* __restrict__ D) {
    const int lane = threadIdx.x & 31;
    const int wave = threadIdx.x >> 5;
    const int tile = blockIdx.x * 8 + wave;       // 65536 tiles total
    const int ti   = tile >> 8;                   // tile row (0..255)
    const int tj   = tile & 255;                  // tile col (0..255)
    const int row0 = ti * 16;
    const int col0 = tj * 16;
    const int l    = lane & 15;
    const int half = lane >> 4;                   // 0: K 0/1, 1: K 2/3 (per chunk)

    const float* ya = y  + (size_t)(row0 + l) * FEAT;
    const float* yb = yh + (size_t)(col0 + l) * FEAT;

    v8f acc = {};
#pragma unroll
    for (int c = 0; c < 4; ++c) {
        const int kb = 4 * c + 2 * half;
        v2f a, b;
        a.x = ya[kb];     a.y = ya[kb + 1];
        b.x = yb[kb];     b.y = yb[kb + 1];
        acc = __builtin_amdgcn_wmma_f32_16x16x4_f32(
                /*neg_a=*/false, a, /*neg_b=*/false, b,
                /*c_mod=*/(short)0, acc, /*reuse_a=*/false, /*reuse_b=*/false);
    }

    // C/D layout: VGPR r -> row (row0 + r + 8*half), col (col0 + l)
    const float nb    = nyh[col0 + l];
    const int   rbase = row0 + 8 * half;
#pragma unroll
    for (int r = 0; r < 8; ++r) {
        float dval = (ny[rbase + r] + nb - 2.0f * acc[r]) * (1.0f / 16.0f);
        dval = fmaxf(dval, 0.0f);                 // guard fp expansion negatives
        D[(size_t)(rbase + r) * HN + (col0 + l)] = dval;
    }
}

// ---------------------------------------------------------------------------
// Kernel 2: serial anti-diagonal DP, single workgroup.
//   A_k[i] = D[i][k-i] + min(A_{k-2}[i-1], A_{k-1}[i-1], A_{k-1}[i])
// Thread t owns rows i = 4t..4t+3; own-row history in registers; only the
// last-row value per thread is exchanged via a 3-slot rotating LDS edge
// buffer (one barrier per diagonal).  D loads for step k+1 are issued before
// the barrier to hide L2 latency behind the sync.
// ---------------------------------------------------------------------------
__global__ __launch_bounds__(NT)
void dtw_dp(const float* __restrict__ D, float* __restrict__ out) {
    const float INF = __builtin_inff();
    const int   t   = threadIdx.x;
    const int   i0  = t * RPT;

    // edge[s][0] is a permanent +inf sentinel (left boundary of thread 0)
    __shared__ float edge[3][NT + 1];
    edge[0][t + 1] = INF;
    edge[1][t + 1] = INF;
    if (t == 0) { edge[0][0] = INF; edge[1][0] = INF; edge[2][0] = INF; }

    float a1[RPT], a2[RPT];                       // A_{k-1}, A_{k-2} (own rows)
#pragma unroll
    for (int r = 0; r < RPT; ++r) { a1[r] = INF; a2[r] = INF; }
    if (t == 0) {
        const float d00 = D[0];
        a2[0] = d00;                              // A_0[0] = D[0][0]
        a1[0] = D[1] + d00;                       // A_1[0] = D[0][1] + D[0][0]
        a1[1] = D[(size_t)HN] + d00;              // A_1[1] = D[1][0] + D[0][0]
    }
    __syncthreads();

    // prefetch D values for k = 2
    float dv[RPT];
#pragma unroll
    for (int r = 0; r < RPT; ++r) {
        const int i = i0 + r, j = 2 - i;
        dv[r] = (j >= 0 && j < HN) ? D[(size_t)i * HN + j] : INF;
    }

    for (int k = 2; k <= 2 * HN - 2; ++k) {
        const int sN = k % 3;
        const int s1 = (k + 2) % 3;               // (k-1) % 3
        const int s2 = (k + 1) % 3;               // (k-2) % 3

        const float e1 = edge[s1][t];             // A_{k-1}[i0-1]
        const float e2 = edge[s2][t];             // A_{k-2}[i0-1]

        float a0[RPT];
        a0[0] = dv[0] + fminf(fminf(e2, e1), a1[0]);
#pragma unroll
        for (int r = 1; r < RPT; ++r)
            a0[r] = dv[r] + fminf(fminf(a2[r - 1], a1[r - 1]), a1[r]);

        edge[sN][t + 1] = a0[RPT - 1];            // publish last row to neighbor

#pragma unroll
        for (int r = 0; r < RPT; ++r) { a2[r] = a1[r]; a1[r] = a0[r]; }

        // issue next step's D loads before the barrier (latency hiding)
        const int kn = k + 1;
#pragma unroll
        for (int r = 0; r < RPT; ++r) {
            const int i = i0 + r, j = kn - i;
            dv[r] = (j >= 0 && j < HN) ? D[(size_t)i * HN + j] : INF;
        }
        __syncthreads();
    }

    if (t == NT - 1) out[0] = a1[RPT - 1];        // cost(4095, 4095)
}

// ---------------------------------------------------------------------------
extern "C" void kernel_launch(void* const* d_in, const int* in_sizes, int n_in,
                              void* d_out, int out_size, void* d_ws, size_t ws_size,
                              hipStream_t stream) {
    const float* y  = (const float*)d_in[0];   // (4096, 16) f32
    const float* yh = (const float*)d_in[1];   // (4096, 16) f32

    float* D   = (float*)d_ws;                                        // 64 MB
    float* ny  = (float*)((char*)d_ws + (size_t)HN * HN * sizeof(float));
    float* nyh = ny + HN;

    dtw_norms<<<HN / 256, 256, 0, stream>>>(y, yh, ny, nyh);
    dtw_dist_wmma<<<(HN / 16) * (HN / 16) / 8, 256, 0, stream>>>(y, yh, ny, nyh, D);
    dtw_dp<<<1, NT, 0, stream>>>(D, (float*)d_out);
}